// MemoryBank_53944789238174
// MI455X (gfx1250) — compile-verified
//
#include <hip/hip_runtime.h>

typedef float v2f __attribute__((ext_vector_type(2)));
typedef float v4f __attribute__((ext_vector_type(4)));
typedef float v8f __attribute__((ext_vector_type(8)));
typedef int   v4i __attribute__((vector_size(16)));   // matches builtin param type

#define DDIM   640
#define NWAY   5
#define NSHOT  5
#define NPAIR  (NWAY * NSHOT)          // 25
#define NMEM   16384
#define STOT   (NMEM + NSHOT)          // 16389
#define TOPK   8
#define EPSN   1e-12f

// workspace layout (in floats)
#define WS_NS   0                       // 25 x 640 normalized support
#define WS_P    (NPAIR * DDIM)          // 5 x 640 mean normalized support
#define WS_SIM  (WS_P + NWAY * DDIM)    // 5 x 16389 similarities

#define KC              32              // K chunk per LDS stage
#define NCHUNK          (DDIM / KC)     // 20
#define ROWS_PER_WAVE   16
#define WAVES           8
#define ROWS_PER_BLOCK  (ROWS_PER_WAVE * WAVES)       // 128
#define F4_PER_LANE     (ROWS_PER_WAVE * KC / 4 / 32) // 4 float4 per lane per chunk

// ---- CDNA5 async global->LDS path (guarded; falls back to NT b128 loads) ----
#if __has_builtin(__builtin_amdgcn_global_load_async_to_lds_b128)
#define ASYNC_LDS 1
#else
#define ASYNC_LDS 0
#endif

#if ASYNC_LDS
#if __has_builtin(__builtin_amdgcn_s_wait_asynccnt)
#define ASYNC_WAIT(nn) __builtin_amdgcn_s_wait_asynccnt(nn)
#else
#define ASYNC_WAIT(nn) asm volatile("s_wait_asynccnt %0" ::"i"(nn) : "memory")
#endif
// builtin signature (from clang diagnostic): arg0 = int4 addrspace(1)*,
// arg1 = int4 addrspace(3)*, then two int immediates (offset, cpol)
#define AS1P(p) ((__attribute__((address_space(1))) v4i*)(p))
#define AS3P(p) ((__attribute__((address_space(3))) v4i*)(p))
#endif

// ---------------------------------------------------------------------------
// Kernel 1: normalize support, P[w] = mean_s nsup[w,s], support part of sim.
// (mean-of-dots == dot-with-mean collapses the GEMM to 5 output columns)
// ---------------------------------------------------------------------------
__global__ void mb_prep_kernel(const float* __restrict__ support,
                               float* __restrict__ ws) {
    float* NS  = ws + WS_NS;
    float* P   = ws + WS_P;
    float* SIM = ws + WS_SIM;

    const int tid  = threadIdx.x;
    const int lane = tid & 31;
    const int wv   = tid >> 5;

    for (int p = wv; p < NPAIR; p += WAVES) {
        const float* src = support + (size_t)p * DDIM;
        float acc = 0.f;
        for (int k = lane; k < DDIM; k += 32) { float x = src[k]; acc += x * x; }
        for (int off = 16; off > 0; off >>= 1) acc += __shfl_xor(acc, off, 32);
        float inv = 1.0f / fmaxf(sqrtf(acc), EPSN);
        for (int k = lane; k < DDIM; k += 32) NS[(size_t)p * DDIM + k] = src[k] * inv;
    }
    __syncthreads();

    for (int idx = tid; idx < NWAY * DDIM; idx += 256) {
        int w = idx / DDIM, k = idx - w * DDIM;
        float s = 0.f;
        for (int sh = 0; sh < NSHOT; ++sh) s += NS[(size_t)(sh * NWAY + w) * DDIM + k];
        P[(size_t)w * DDIM + k] = s * (1.0f / NSHOT);
    }
    __syncthreads();

    for (int p = wv; p < NPAIR; p += WAVES) {
        int w = p / NSHOT, j = p - w * NSHOT;
        float acc = 0.f;
        for (int k = lane; k < DDIM; k += 32)
            acc += P[(size_t)w * DDIM + k] * NS[(size_t)(j * NWAY + w) * DDIM + k];
        for (int off = 16; off > 0; off >>= 1) acc += __shfl_xor(acc, off, 32);
        if (lane == 0) SIM[(size_t)w * STOT + j] = acc;
    }
}

// ---------------------------------------------------------------------------
// Kernel 2: HBM-stream pass over memory_encoded (the whole cost of this op).
// Per wave: 16 rows. Double-buffered async global->LDS b128 copies overlap
// the v_wmma_f32_16x16x4_f32 accumulation; row L2-norms are fused onto the
// WMMA A-fragment reads (lane n + lane n+16 together cover row n exactly).
// ---------------------------------------------------------------------------
__global__ void mb_sim_kernel(const float* __restrict__ mem,
                              float* __restrict__ ws) {
    const float* Pg  = ws + WS_P;
    float*       SIM = ws + WS_SIM;

    __shared__ float Pl[DDIM * NWAY];                       // swizzled (k>>1)*10 + w*2 + (k&1)
    __shared__ float As[2][WAVES][ROWS_PER_WAVE * KC];      // double-buffered A staging
    __shared__ float Nrm[WAVES][16];                        // per-row sum of squares

    const int tid  = threadIdx.x;
    const int lane = tid & 31;
    const int wv   = tid >> 5;
    const int n    = lane & 15;       // WMMA M (A) / N (B) index
    const int half = lane >> 4;
    const int koff = half * 2;        // K-pair slot within the x4 step

    // P tile into LDS, swizzled so (k, k+1) of a column are adjacent (float2)
    for (int idx = tid; idx < DDIM * NWAY; idx += 256) {
        int k = idx / NWAY, w = idx - k * NWAY;
        Pl[(k >> 1) * (NWAY * 2) + w * 2 + (k & 1)] = Pg[(size_t)w * DDIM + k];
    }
    __syncthreads();

    const int rowbase = blockIdx.x * ROWS_PER_BLOCK + wv * ROWS_PER_WAVE;

    v8f   acc  = {};
    float nsqp = 0.f;   // partial sum-of-squares of row n (this half's K slots)

    auto stage = [&](int c, int buf) {
#pragma unroll
        for (int i = 0; i < F4_PER_LANE; ++i) {
            int u   = i * 32 + lane;
            int row = u >> 3;          // KC/4 = 8 float4 per row
            int c4  = u & 7;
            const float* gp = mem + (size_t)(rowbase + row) * DDIM + c * KC + c4 * 4;
            float*       lp = &As[buf][wv][row * KC + c4 * 4];
#if ASYNC_LDS
            __builtin_amdgcn_global_load_async_to_lds_b128(AS1P(gp), AS3P(lp), 0, 0);
#else
            *(v4f*)lp = __builtin_nontemporal_load((const v4f*)gp);
#endif
        }
    };

    auto consume = [&](int c, int buf) {
#pragma unroll
        for (int k4 = 0; k4 < KC; k4 += 4) {
            v2f a = *(const v2f*)&As[buf][wv][n * KC + k4 + koff];
            nsqp += a.x * a.x + a.y * a.y;          // fused row-norm on the fragment
            v2f b = {0.f, 0.f};
            if (n < NWAY) {
                int kg = c * KC + k4 + koff;        // even by construction
                b = *(const v2f*)&Pl[(kg >> 1) * (NWAY * 2) + n * 2];
            }
            acc = __builtin_amdgcn_wmma_f32_16x16x4_f32(
                    false, a, false, b, (short)0, acc, false, false);
        }
    };

#if ASYNC_LDS
    // software pipeline: copy of chunk c+1 overlaps WMMA of chunk c;
    // per-wave LDS regions mean ASYNCcnt alone orders producer/consumer.
    stage(0, 0);
    for (int c = 0; c < NCHUNK; ++c) {
        if (c + 1 < NCHUNK) {
            stage(c + 1, (c + 1) & 1);
            ASYNC_WAIT(F4_PER_LANE);   // chunk c's copies (older) are done
        } else {
            ASYNC_WAIT(0);
        }
        consume(c, c & 1);
    }
#else
    for (int c = 0; c < NCHUNK; ++c) {
        stage(c, 0);
        __syncthreads();
        consume(c, 0);
        __syncthreads();
    }
#endif

    // combine the two half-wave K slots -> full ||row n||^2, exchange via LDS
    nsqp += __shfl_xor(nsqp, 16, 32);
    if (lane < 16) Nrm[wv][lane] = nsqp;
    __syncthreads();

    // C/D layout: lane column N = n; VGPR vr holds M = vr + half*8
    if (n < NWAY) {
#pragma unroll
        for (int vr = 0; vr < 8; ++vr) {
            int m = vr + half * 8;
            float invn = 1.0f / fmaxf(sqrtf(Nrm[wv][m]), EPSN);
            SIM[(size_t)n * STOT + NSHOT + rowbase + m] = acc[vr] * invn;
        }
    }
}

// ---------------------------------------------------------------------------
// Kernel 3: per-way top-8 over S=16389 sims, weighted average of raw vectors.
// ---------------------------------------------------------------------------
__global__ void mb_proto_kernel(const float* __restrict__ support,
                                const float* __restrict__ mem,
                                const float* __restrict__ sim_all,
                                float* __restrict__ out) {
    __shared__ float cv[256 * TOPK];
    __shared__ int   ci[256 * TOPK];
    __shared__ float tv[TOPK];
    __shared__ int   ti[TOPK];

    const int way = blockIdx.x;
    const int tid = threadIdx.x;
    const float* sim = sim_all + (size_t)way * STOT;

    float bv[TOPK]; int bi[TOPK];
#pragma unroll
    for (int j = 0; j < TOPK; ++j) { bv[j] = -3.4e38f; bi[j] = -1; }
    for (int i = tid; i < STOT; i += 256) {
        float v = sim[i];
        if (v > bv[TOPK - 1]) {
            int j = TOPK - 1;
            while (j > 0 && bv[j - 1] < v) { bv[j] = bv[j - 1]; bi[j] = bi[j - 1]; --j; }
            bv[j] = v; bi[j] = i;
        }
    }
#pragma unroll
    for (int j = 0; j < TOPK; ++j) { cv[tid * TOPK + j] = bv[j]; ci[tid * TOPK + j] = bi[j]; }
    __syncthreads();

    if (tid == 0) {
        float fv[TOPK]; int fi[TOPK];
        for (int j = 0; j < TOPK; ++j) { fv[j] = -3.4e38f; fi[j] = -1; }
        for (int i = 0; i < 256 * TOPK; ++i) {
            float v = cv[i]; int idx = ci[i];
            if (idx >= 0 && v > fv[TOPK - 1]) {
                int j = TOPK - 1;
                while (j > 0 && fv[j - 1] < v) { fv[j] = fv[j - 1]; fi[j] = fi[j - 1]; --j; }
                fv[j] = v; fi[j] = idx;
            }
        }
        for (int j = 0; j < TOPK; ++j) { tv[j] = fv[j]; ti[j] = fi[j]; }
    }
    __syncthreads();

    float denom = 0.f;
#pragma unroll
    for (int j = 0; j < TOPK; ++j) denom += tv[j];

    for (int k = tid; k < DDIM; k += 256) {
        float num = 0.f;
#pragma unroll
        for (int j = 0; j < TOPK; ++j) {
            int idx = ti[j];
            const float* vec = (idx < NSHOT)
                ? (support + ((size_t)idx * NWAY + way) * DDIM)   // raw support shot
                : (mem + (size_t)(idx - NSHOT) * DDIM);           // raw memory vector
            num += tv[j] * vec[k];
        }
        out[(size_t)way * DDIM + k] = num / denom;
    }
}

// ---------------------------------------------------------------------------
extern "C" void kernel_launch(void* const* d_in, const int* in_sizes, int n_in,
                              void* d_out, int out_size, void* d_ws, size_t ws_size,
                              hipStream_t stream) {
    (void)in_sizes; (void)n_in; (void)out_size; (void)ws_size;
    const float* support = (const float*)d_in[0];   // [1,5,5,640]
    const float* mem     = (const float*)d_in[1];   // [16384,640]
    float*       ws      = (float*)d_ws;
    float*       out     = (float*)d_out;           // [1,5,640]

    mb_prep_kernel<<<1, 256, 0, stream>>>(support, ws);
    mb_sim_kernel<<<NMEM / ROWS_PER_BLOCK, 256, 0, stream>>>(mem, ws);
    mb_proto_kernel<<<NWAY, 256, 0, stream>>>(support, mem, ws + WS_SIM, out);
}